// GraphAttentionLayer_1580547974609
// MI455X (gfx1250) — compile-verified
//
#include <hip/hip_runtime.h>
#include <hip/hip_bf16.h>

#define IN_F   128
#define OUT_F  64
#define BATCH  4
#define NN     2048
#define ALPHA  0.2f
#define NEG_INF (-9000000000000000.0f)
#define TILE_M 16
// 2048 % 64banks == 0 -> pad row stride by 4 floats so A-fragment column
// reads (stride LDS_STRIDE) hit distinct banks: 2052 % 64 = 4.
#define LDS_STRIDE 2052

typedef __attribute__((ext_vector_type(2))) float v2f;
typedef __attribute__((ext_vector_type(8))) float v8f;

// ---------------------------------------------------------------------------
// Kernel A: Wh = h @ W  (per-row 128x64 GEMV), plus e_l = Wh.a_l, e_r = Wh.a_r
// 8192 blocks x 64 threads; memory-trivial (h is 4 MB, W/a tiny).
// ---------------------------------------------------------------------------
__global__ void __launch_bounds__(64)
wh_el_er_kernel(const float* __restrict__ h, const float* __restrict__ W,
                const float* __restrict__ a, float* __restrict__ Wh,
                float* __restrict__ e_l, float* __restrict__ e_r) {
  __shared__ float hrow[IN_F];
  __shared__ float red_l[64];
  __shared__ float red_r[64];
  const int row = blockIdx.x;   // 0 .. B*N-1
  const int o   = threadIdx.x;  // 0 .. 63
  hrow[o]      = h[(size_t)row * IN_F + o];
  hrow[o + 64] = h[(size_t)row * IN_F + o + 64];
  __syncthreads();
  float acc = 0.f;
#pragma unroll 8
  for (int f = 0; f < IN_F; ++f)
    acc = fmaf(hrow[f], W[f * OUT_F + o], acc);
  Wh[(size_t)row * OUT_F + o] = acc;
  red_l[o] = acc * a[o];            // a_l = a[0:64]
  red_r[o] = acc * a[OUT_F + o];    // a_r = a[64:128]
  __syncthreads();
  for (int s = 32; s > 0; s >>= 1) {
    if (o < s) { red_l[o] += red_l[o + s]; red_r[o] += red_r[o + s]; }
    __syncthreads();
  }
  if (o == 0) { e_l[row] = red_l[0]; e_r[row] = red_r[0]; }
}

// ---------------------------------------------------------------------------
// Kernel B (fused): for a 16-row tile of one batch:
//   Phase 1: e = leaky(e_l[i]+e_r[j]), mask by adj, row-softmax -> LDS tile
//            (128 KB of the WGP's 320 KB LDS) and stream to d_out.attention.
//   Phase 2: h_prime tile = attn_tile(16x2048) @ Wh(2048x64) via fp32 WMMA
//            V_WMMA_F32_16X16X4_F32 (exact fp32, K-step 4), + leaky(0.01).
// Block = 256 threads = 8 wave32; waves 0..7 own 2 softmax rows each;
// waves 0..3 own 16-wide N-tiles for the WMMA GEMM.
// ---------------------------------------------------------------------------
__global__ void __launch_bounds__(256)
gat_attn_kernel(const int* __restrict__ adj, const float* __restrict__ Wh,
                const float* __restrict__ e_l, const float* __restrict__ e_r,
                float* __restrict__ h_prime, float* __restrict__ attn) {
  extern __shared__ float tile[];   // [TILE_M][LDS_STRIDE]
  const int tid   = threadIdx.x;
  const int lane  = tid & 31;
  const int wave  = tid >> 5;                     // 0..7
  const int b     = blockIdx.x / (NN / TILE_M);
  const int i_base = (blockIdx.x % (NN / TILE_M)) * TILE_M;

  const float* er_row = e_r + (size_t)b * NN;

  // ---- Phase 1: masked leaky scores + softmax, one wave per 2 rows ----
  for (int rr = 0; rr < 2; ++rr) {
    const int r = wave * 2 + rr;
    const int i = i_base + r;
    const float el = e_l[(size_t)b * NN + i];
    const int* adj_row = adj + ((size_t)b * NN + i) * NN;
    float* trow = tile + r * LDS_STRIDE;

    float vmax = -3.4e38f;
    for (int j = lane; j < NN; j += 32) {
      float x = el + er_row[j];
      x = x > 0.f ? x : ALPHA * x;               // leaky_relu(. , 0.2)
      float v = adj_row[j] > 0 ? x : NEG_INF;    // mask
      trow[j] = v;
      vmax = fmaxf(vmax, v);
    }
    for (int off = 16; off > 0; off >>= 1)
      vmax = fmaxf(vmax, __shfl_xor(vmax, off, 32));

    float vsum = 0.f;
    for (int j = lane; j < NN; j += 32) {
      float ex = __expf(trow[j] - vmax);         // all-masked row -> exp(0)=1
      trow[j] = ex;
      vsum += ex;
    }
    for (int off = 16; off > 0; off >>= 1)
      vsum += __shfl_xor(vsum, off, 32);

    const float inv = 1.f / vsum;
    float* arow = attn + ((size_t)b * NN + i) * NN;
    for (int j = lane; j < NN; j += 32) {
      float p = trow[j] * inv;
      trow[j] = p;        // keep normalized tile in LDS for the GEMM
      arow[j] = p;        // the one mandatory HBM write of attention
    }
  }
  __syncthreads();

  // ---- Phase 2: 16x64 output tile = tile(16x2048) @ Wh(2048x64) ----
  if (wave < 4) {
    const int n_base = wave * TILE_M;
    const int hi = (lane >> 4) & 1;   // lanes 16..31 hold K+2 / M+8 halves
    const int m  = lane & 15;
    const float* whb = Wh + (size_t)b * NN * OUT_F;

    v8f c = {};
    for (int k = 0; k < NN; k += 4) {
      // A 16x4 f32 frag: lane(m,hi) -> VGPR0=A[m][k+2hi], VGPR1=A[m][k+2hi+1]
      const float* ap = tile + (size_t)m * LDS_STRIDE + k + 2 * hi;
      v2f av; av.x = ap[0]; av.y = ap[1];
      // B 4x16 f32 frag: lane(n=m,hi) -> VGPR0=B[2hi][n], VGPR1=B[2hi+1][n]
      const float* bp = whb + (size_t)(k + 2 * hi) * OUT_F + n_base + m;
      v2f bv; bv.x = bp[0]; bv.y = bp[OUT_F];
      c = __builtin_amdgcn_wmma_f32_16x16x4_f32(false, av, false, bv,
                                                (short)0, c, false, false);
    }
    // C/D layout: VGPR r -> row M = r + 8*hi, col N = lane&15
    float* hp = h_prime + ((size_t)b * NN + i_base) * OUT_F;
#pragma unroll
    for (int r = 0; r < 8; ++r) {
      float v = c[r];
      v = v > 0.f ? v : 0.01f * v;               // leaky_relu(. , 0.01)
      hp[(size_t)(r + 8 * hi) * OUT_F + n_base + m] = v;
    }
  }
}

// ---------------------------------------------------------------------------
extern "C" void kernel_launch(void* const* d_in, const int* in_sizes, int n_in,
                              void* d_out, int out_size, void* d_ws, size_t ws_size,
                              hipStream_t stream) {
  (void)in_sizes; (void)n_in; (void)out_size; (void)ws_size;
  const float* h   = (const float*)d_in[0];   // (4,2048,128) f32
  const int*   adj = (const int*)d_in[1];     // (4,2048,2048) i32
  const float* W   = (const float*)d_in[2];   // (128,64) f32
  const float* a   = (const float*)d_in[3];   // (128,1) f32

  float* out     = (float*)d_out;
  float* h_prime = out;                                   // (4,2048,64)
  float* attn    = out + (size_t)BATCH * NN * OUT_F;      // (4,2048,2048)

  float* Wh  = (float*)d_ws;                              // 2 MB
  float* e_l = Wh + (size_t)BATCH * NN * OUT_F;           // 32 KB
  float* e_r = e_l + (size_t)BATCH * NN;                  // 32 KB

  wh_el_er_kernel<<<BATCH * NN, 64, 0, stream>>>(h, W, a, Wh, e_l, e_r);

  const size_t lds_bytes = (size_t)TILE_M * LDS_STRIDE * sizeof(float); // 131328
  gat_attn_kernel<<<BATCH * (NN / TILE_M), 256, lds_bytes, stream>>>(
      adj, Wh, e_l, e_r, h_prime, attn);
}